// GraphAttentionLayer_17128329576568
// MI455X (gfx1250) — compile-verified
//
#include <hip/hip_runtime.h>

#define NN   8192
#define INF_ 512
#define OF   256
#define HP   264   // padded LDS row stride (floats): 2*HP % 64 == 16 banks apart

typedef __attribute__((ext_vector_type(2))) float v2f;
typedef __attribute__((ext_vector_type(8))) float v8f;

// D = A(16x4) * B(4x16) + C, full f32 WMMA (CDNA5)
__device__ __forceinline__ v8f wmma4(v2f a, v2f b, v8f c) {
  return __builtin_amdgcn_wmma_f32_16x16x4_f32(false, a, false, b, (short)0, c,
                                               false, false);
}

// async global -> LDS copy, 16 bytes per lane, tracked by ASYNCcnt
__device__ __forceinline__ void async_b128(unsigned lds_addr, const float* g) {
  asm volatile("global_load_async_to_lds_b128 %0, %1, off"
               :: "v"(lds_addr), "v"(g)
               : "memory");
}
__device__ __forceinline__ void wait_async0() {
  asm volatile("s_wait_asynccnt 0x0" ::: "memory");
}

// mish(x) = x * tanh(softplus(x)) = x * (t^2 + 2t) / (t^2 + 2t + 2), t = e^x
__device__ __forceinline__ float mish_f(float x) {
  float t = __expf(x);
  float u = t * (t + 2.0f);
  float r = x * u * __builtin_amdgcn_rcpf(u + 2.0f);
  return (x > 20.0f) ? x : r;  // avoid inf/inf for large x
}

// ---------------- Kernel 1: h = input @ W  (8192x512 @ 512x256) ------------
__global__ __launch_bounds__(256) void k_gemm_hw(const float* __restrict__ in,
                                                 const float* __restrict__ W,
                                                 float* __restrict__ h) {
  const int tid  = threadIdx.x;
  const int lane = tid & 31;
  const int wave = tid >> 5;
  const int r  = lane & 15;
  const int hr = lane >> 4;
  const int i0 = (blockIdx.x * 8 + wave) * 16;

  const float* Ar = in + (size_t)(i0 + r) * INF_ + 2 * hr;
  const float* Wl = W + (size_t)(2 * hr) * OF + r;

  v8f acc[16] = {};
  for (int k = 0; k < INF_; k += 4) {
    v2f a = *(const v2f*)(Ar + k);
#pragma unroll
    for (int t = 0; t < 16; ++t) {
      v2f b;
      b.x = Wl[(size_t)k * OF + 16 * t];
      b.y = Wl[(size_t)(k + 1) * OF + 16 * t];
      acc[t] = wmma4(a, b, acc[t]);
    }
  }
#pragma unroll
  for (int t = 0; t < 16; ++t)
#pragma unroll
    for (int vr = 0; vr < 8; ++vr)
      h[(size_t)(i0 + vr + 8 * hr) * OF + 16 * t + r] = acc[t][vr];
}

// ---------------- Kernel 2: s1 = h @ a[:256], s2 = h @ a[256:] -------------
__global__ __launch_bounds__(256) void k_scores(const float* __restrict__ h,
                                                const float* __restrict__ a,
                                                float* __restrict__ s1,
                                                float* __restrict__ s2) {
  const int lane = threadIdx.x & 31;
  const int wave = threadIdx.x >> 5;
  const int row  = blockIdx.x * 8 + wave;
  const float* hr = h + (size_t)row * OF;
  float acc1 = 0.f, acc2 = 0.f;
#pragma unroll
  for (int q = 0; q < 8; ++q) {
    float hv = hr[lane + 32 * q];
    acc1 += hv * a[lane + 32 * q];
    acc2 += hv * a[OF + lane + 32 * q];
  }
#pragma unroll
  for (int off = 16; off > 0; off >>= 1) {
    acc1 += __shfl_xor(acc1, off, 32);
    acc2 += __shfl_xor(acc2, off, 32);
  }
  if (lane == 0) { s1[row] = acc1; s2[row] = acc2; }
}

// ------ Kernel 3: fused masked mish-softmax + attn @ h + mish epilogue -----
// Block = 256 thr = 8 waves: 4 i-tiles (16 rows) x 2 N-halves (128 cols).
// Even wave of each pair computes the softmax tile; both waves do WMMA.
__global__ __launch_bounds__(256) void k_attn(const float* __restrict__ adj,
                                              const float* __restrict__ h,
                                              const float* __restrict__ s1,
                                              const float* __restrict__ s2,
                                              float* __restrict__ out) {
  __shared__ float hb[2][16][HP];   // double-buffered h tile, row-major
  __shared__ v2f   pbuf[4][32][4];  // p tile in WMMA-A per-lane layout
  __shared__ float fbuf[4][16];     // per-row rescale factor
  __shared__ float rbuf[4][16];     // per-row exp sum

  const int tid  = threadIdx.x;
  const int lane = tid & 31;
  const int wave = tid >> 5;
  const int r  = lane & 15;
  const int hr = lane >> 4;
  const int it = wave >> 1;               // i-tile within block
  const int i0 = blockIdx.x * 64 + it * 16;
  const int nb = (wave & 1) * 128;        // N-half base
  const int row = i0 + r;

  const float* adjRow = adj + (size_t)row * NN;
  const float s1r = s1[row];

  // staging thread mapping: 16 threads per h row, 16 cols each
  const int sr = tid >> 4;
  const int sc = (tid & 15) * 16;

  float m = -3.0e38f;
  float l = 0.0f;
  v8f acc[8] = {};

  // prologue: stream tile 0 into buffer 0
  {
    const float* g = h + (size_t)sr * OF + sc;
#pragma unroll
    for (int q = 0; q < 4; ++q)
      async_b128((unsigned)(uintptr_t)&hb[0][sr][sc + 4 * q], g + 4 * q);
  }
  wait_async0();
  __syncthreads();

  for (int j0 = 0; j0 < NN; j0 += 16) {
    const int cur = (j0 >> 4) & 1;
    const int nj  = j0 + 16;
    if (nj < NN) {  // stream next tile into the spare buffer
      const float* g = h + (size_t)(nj + sr) * OF + sc;
#pragma unroll
      for (int q = 0; q < 4; ++q)
        async_b128((unsigned)(uintptr_t)&hb[cur ^ 1][sr][sc + 4 * q], g + 4 * q);
    }

    v2f p[4];
    if ((wave & 1) == 0) {
      if (j0 + 256 < NN) __builtin_prefetch(adjRow + j0 + 256, 0, 0);
      // masked e values for this 16x16 tile, WMMA-A layout
      float e[8];
      float tmax = -3.0e38f;
#pragma unroll
      for (int c = 0; c < 4; ++c) {
        int jj = j0 + 4 * c + 2 * hr;
        float2 av = *(const float2*)(adjRow + jj);
        float2 sv = *(const float2*)(s2 + jj);
        float e0 = (av.x > 0.f) ? mish_f(s1r + sv.x) : -9.0e15f;
        float e1 = (av.y > 0.f) ? mish_f(s1r + sv.y) : -9.0e15f;
        e[2 * c] = e0; e[2 * c + 1] = e1;
        tmax = fmaxf(tmax, fmaxf(e0, e1));
      }
      tmax = fmaxf(tmax, __shfl_xor(tmax, 16, 32));  // combine half-rows
      float mnew = fmaxf(m, tmax);
      float f = __expf(m - mnew);
      float rsum = 0.f;
#pragma unroll
      for (int c = 0; c < 4; ++c) {
        float p0 = __expf(e[2 * c] - mnew);      // masked -> underflow to 0
        float p1 = __expf(e[2 * c + 1] - mnew);
        p[c].x = p0; p[c].y = p1;
        rsum += p0 + p1;
        pbuf[it][lane][c] = p[c];
      }
      rsum += __shfl_xor(rsum, 16, 32);
      m = mnew;
      if (hr == 0) { fbuf[it][r] = f; rbuf[it][r] = rsum; }
    }
    __syncthreads();

    if (wave & 1) {
#pragma unroll
      for (int c = 0; c < 4; ++c) p[c] = pbuf[it][lane][c];
    }
    float f = fbuf[it][r];
    l = l * f + rbuf[it][r];

    // rescale accumulators: per-C-row factors broadcast from LDS
    float fr[8];
#pragma unroll
    for (int vr = 0; vr < 8; ++vr) fr[vr] = fbuf[it][vr + 8 * hr];
#pragma unroll
    for (int t = 0; t < 8; ++t)
#pragma unroll
      for (int vr = 0; vr < 8; ++vr) acc[t][vr] *= fr[vr];

    // acc += P(16x16) @ Htile(16x128-half): 4 K-chunks x 8 N-tiles
#pragma unroll
    for (int c = 0; c < 4; ++c) {
#pragma unroll
      for (int t = 0; t < 8; ++t) {
        v2f b;
        b.x = hb[cur][4 * c + 2 * hr + 0][nb + 16 * t + r];
        b.y = hb[cur][4 * c + 2 * hr + 1][nb + 16 * t + r];
        acc[t] = wmma4(p[c], b, acc[t]);
      }
    }
    wait_async0();   // next tile fully landed in LDS
    __syncthreads(); // everyone done with cur buffer + p tile
  }

  // normalize + mish epilogue
  float inv = 1.0f / l;
  float ir[8];
#pragma unroll
  for (int vr = 0; vr < 8; ++vr) ir[vr] = __shfl(inv, vr + 8 * hr, 32);
#pragma unroll
  for (int t = 0; t < 8; ++t)
#pragma unroll
    for (int vr = 0; vr < 8; ++vr)
      out[(size_t)(i0 + vr + 8 * hr) * OF + nb + 16 * t + r] =
          mish_f(acc[t][vr] * ir[vr]);
}

extern "C" void kernel_launch(void* const* d_in, const int* in_sizes, int n_in,
                              void* d_out, int out_size, void* d_ws,
                              size_t ws_size, hipStream_t stream) {
  const float* input = (const float*)d_in[0];  // 8192x512
  const float* adj   = (const float*)d_in[1];  // 8192x8192
  const float* W     = (const float*)d_in[2];  // 512x256
  const float* a     = (const float*)d_in[3];  // 512x1
  float* out = (float*)d_out;                  // 8192x256

  float* h  = (float*)d_ws;                    // 8192*256 f32 = 8 MB
  float* s1 = h + (size_t)NN * OF;
  float* s2 = s1 + NN;

  k_gemm_hw<<<NN / 16 / 8, 256, 0, stream>>>(input, W, h);
  k_scores<<<NN / 8, 256, 0, stream>>>(h, a, s1, s2);
  k_attn<<<NN / 64, 256, 0, stream>>>(adj, h, s1, s2, out);
}